// HighOrderActivationB_89446988906950
// MI455X (gfx1250) — compile-verified
//
#include <hip/hip_runtime.h>

// HighOrderActivationB — bandwidth-bound gather/weighted-sum.
// B=4096, G=1024, r=3, T=27, D=8, ACT_FACTOR=1.0 (identity).
// HBM floor: 48 MiB (X in) + 128 MiB (out) => ~7.6 us at 23.3 TB/s.
// params (864 KB total) is L2-resident; per-block slice staged in LDS via
// CDNA5 GLOBAL_LOAD_ASYNC_TO_LDS_B128 + s_wait_asynccnt.
// LDS group stride padded 216 -> 220 floats so half-wave ds_load_b128
// gathers spread over all 16 four-bank slots (was 8 -> guaranteed 2-way).
// Output written with non-temporal b128 stores (write-once stream).

namespace {
constexpr int BATCH   = 4096;
constexpr int GROUPS  = 1024;
constexpr int ARITY   = 3;
constexpr int TERMS   = 27;           // 3^ARITY
constexpr int OUT_DIM = 8;
constexpr int REF_IND = 13;           // (3^3 - 1)/2

constexpr int GPB     = 16;           // groups per block
constexpr int BPB     = 64;           // batch rows per block
constexpr int THREADS = 256;          // 8 wave32
constexpr int ROWS_PER_ITER = THREADS / GPB;  // 16
constexpr int NITER   = BPB / ROWS_PER_ITER;  // 4

constexpr int GROUP_FLOATS = TERMS * OUT_DIM;        // 216 (contiguous in HBM)
constexpr int GROUP_VEC4   = GROUP_FLOATS / 4;       // 54 float4 per group
constexpr int GROUP_STRIDE = 220;                    // padded LDS stride (floats, 16B-mult)
constexpr int GROUP_STRIDE_VEC4 = GROUP_STRIDE / 4;  // 55
constexpr int TILE_VEC4    = GPB * GROUP_VEC4;       // 864 x 16B transfers
constexpr int LDS_FLOATS   = GPB * GROUP_STRIDE;     // 3520 floats = 14080 B
}

typedef int   v4i __attribute__((ext_vector_type(4)));
typedef float v4f __attribute__((ext_vector_type(4)));
typedef __attribute__((address_space(1))) v4i* gv4p;  // global int4*
typedef __attribute__((address_space(3))) v4i* lv4p;  // LDS int4*

__global__ __launch_bounds__(THREADS) void
hoact_kernel(const float* __restrict__ X,
             const float* __restrict__ P,
             float* __restrict__ out)
{
    __shared__ __align__(16) float sP[LDS_FLOATS];

    const int tid   = threadIdx.x;
    const int gBase = blockIdx.x * GPB;
    const int bBase = blockIdx.y * BPB;

    // ---- Stage params tile (16 groups x 27 x 8 f32) into padded LDS ----
    {
        const v4f* gsrc = (const v4f*)(P + (size_t)gBase * GROUP_FLOATS);
        for (int i = tid; i < TILE_VEC4; i += THREADS) {
            const int grp = i / GROUP_VEC4;              // 0..15
            const int rem = i - grp * GROUP_VEC4;        // 0..53
            const int dst = grp * GROUP_STRIDE_VEC4 + rem;
#if defined(__gfx1250__) && __has_builtin(__builtin_amdgcn_global_load_async_to_lds_b128)
            __builtin_amdgcn_global_load_async_to_lds_b128(
                (gv4p)(const void*)(gsrc + i),
                (lv4p)(void*)(((v4f*)sP) + dst),
                /*offset=*/0, /*cpol=*/0);
#else
            ((v4f*)sP)[dst] = gsrc[i];
#endif
        }
#if defined(__gfx1250__) && __has_builtin(__builtin_amdgcn_global_load_async_to_lds_b128)
#if __has_builtin(__builtin_amdgcn_s_wait_asynccnt)
        __builtin_amdgcn_s_wait_asynccnt(0);
#else
        asm volatile("s_wait_asynccnt 0" ::: "memory");
#endif
#endif
    }
    __syncthreads();

    const int gLocal = tid & (GPB - 1);   // 0..15, lane-consecutive groups
    const int g      = gBase + gLocal;
    const int bLane  = tid >> 4;          // 0..15
    const float* __restrict__ sPg = sP + gLocal * GROUP_STRIDE;

#pragma unroll
    for (int it = 0; it < NITER; ++it) {
        const int b = bBase + bLane + it * ROWS_PER_ITER;
        const float* __restrict__ xp =
            X + (size_t)b * (GROUPS * ARITY) + (size_t)g * ARITY;

        // Prefetch next iteration's X row segment (streams once through HBM).
        if (it + 1 < NITER)
            __builtin_prefetch(xp + (size_t)ROWS_PER_ITER * GROUPS * ARITY, 0, 0);

        const float a0 = xp[0];
        const float a1 = xp[1];
        const float a2 = xp[2];

        // Keys = |a|; payload t_k = sgn(a_k) * 3^k  (sgn(0) = +1, as in reference)
        float k0 = fabsf(a0), k1 = fabsf(a1), k2 = fabsf(a2);
        int   t0 = (a0 >= 0.0f) ? 1 : -1;
        int   t1 = (a1 >= 0.0f) ? 3 : -3;
        int   t2 = (a2 >= 0.0f) ? 9 : -9;

        // Stable ascending sort network: swap only on strict '>' (matches argsort).
        if (k0 > k1) { float tk = k0; k0 = k1; k1 = tk; int tt = t0; t0 = t1; t1 = tt; }
        if (k1 > k2) { float tk = k1; k1 = k2; k2 = tk; int tt = t1; t1 = t2; t2 = tt; }
        if (k0 > k1) { float tk = k0; k0 = k1; k1 = tk; int tt = t0; t0 = t1; t1 = tt; }

        // Suffix sums of payloads -> table indices (each in [0, 26]).
        const int idx2 = REF_IND + t2;
        const int idx1 = idx2 + t1;
        const int idx0 = idx1 + t0;

        // Difference coefficients.
        const float c0 = k0;
        const float c1 = k1 - k0;
        const float c2 = k2 - k1;

        const float* __restrict__ r0 = sPg + idx0 * OUT_DIM;
        const float* __restrict__ r1 = sPg + idx1 * OUT_DIM;
        const float* __restrict__ r2 = sPg + idx2 * OUT_DIM;

        float acc[OUT_DIM];
#pragma unroll
        for (int d = 0; d < OUT_DIM; ++d)
            acc[d] = c0 * r0[d] + c1 * r1[d] + c2 * r2[d];

        // Non-temporal streaming store: out is written once, never re-read.
        v4f lo = { acc[0], acc[1], acc[2], acc[3] };
        v4f hi = { acc[4], acc[5], acc[6], acc[7] };
        v4f* op = (v4f*)(out + (size_t)b * (GROUPS * OUT_DIM) + (size_t)g * OUT_DIM);
        __builtin_nontemporal_store(lo, op + 0);
        __builtin_nontemporal_store(hi, op + 1);
    }
}

extern "C" void kernel_launch(void* const* d_in, const int* in_sizes, int n_in,
                              void* d_out, int out_size, void* d_ws, size_t ws_size,
                              hipStream_t stream) {
    const float* X = (const float*)d_in[0];   // (4096, 3072) f32
    const float* P = (const float*)d_in[1];   // (1024, 27, 8) f32
    float* out = (float*)d_out;               // (4096, 8192) f32

    dim3 grid(GROUPS / GPB, BATCH / BPB);     // 64 x 64 blocks
    hoact_kernel<<<grid, THREADS, 0, stream>>>(X, P, out);
}